// BlockQuantizer_29497835389852
// MI455X (gfx1250) — compile-verified
//
#include <hip/hip_runtime.h>

// SWALP global block-quantize for MI455X (gfx1250, wave32).
// Pass A: abs-max reduce (B128 loads, wave32 shuffle butterfly, LDS tree,
//         one global_atomic_max_u32 per block -- deterministic).
// Pass B: quantize (exact ilogb/ldexp/rndne path, B128 loads, NT B128 stores
//         so the output stream does not evict the ~205MB input from the 192MB L2).
// 32-bit indexing throughout (n comes from an int) -> GVS addressing
// (SGPR64 base + VGPR32 offset), 2x unrolled streaming loops for MLP.

typedef float vf4 __attribute__((ext_vector_type(4)));

#define THREADS 256
#define MAXBLOCKS 4096

__device__ __forceinline__ float wave_max32(float v) {
#pragma unroll
  for (int off = 16; off > 0; off >>= 1)
    v = fmaxf(v, __shfl_xor(v, off, 32));   // wave32 butterfly (ds_swizzle path)
  return v;
}

__device__ __forceinline__ float absmax4(vf4 v) {
  return fmaxf(fmaxf(__builtin_fabsf(v.x), __builtin_fabsf(v.y)),
               fmaxf(__builtin_fabsf(v.z), __builtin_fabsf(v.w)));
}

__global__ void swalp_init(unsigned* __restrict__ ws) {
  if (threadIdx.x == 0) ws[0] = 0u;
}

__global__ __launch_bounds__(THREADS)
void swalp_absmax(const vf4* __restrict__ x4, const float* __restrict__ x,
                  unsigned n4, unsigned n, unsigned* __restrict__ ws) {
  float m = 0.0f;
  const unsigned stride = gridDim.x * THREADS;
  unsigned i = blockIdx.x * THREADS + threadIdx.x;
  // 2x unrolled main stream: two independent B128 loads in flight per trip
  for (; i + stride < n4; i += 2u * stride) {
    vf4 a = x4[i];
    vf4 b = x4[i + stride];
    m = fmaxf(m, fmaxf(absmax4(a), absmax4(b)));
  }
  if (i < n4) m = fmaxf(m, absmax4(x4[i]));
  if (blockIdx.x == 0 && threadIdx.x == 0)                  // tail (n % 4), generic
    for (unsigned t = n4 * 4u; t < n; ++t) m = fmaxf(m, __builtin_fabsf(x[t]));

  m = wave_max32(m);
  __shared__ float smax[THREADS / 32];
  const int lane = threadIdx.x & 31;
  const int wv   = threadIdx.x >> 5;
  if (lane == 0) smax[wv] = m;
  __syncthreads();
  if (wv == 0) {
    m = (lane < THREADS / 32) ? smax[lane] : 0.0f;
    m = wave_max32(m);
    // abs values are >= 0: unsigned bit-pattern order == float order.
    if (lane == 0) atomicMax(ws, __float_as_uint(m));       // global_atomic_max_u32
  }
}

__device__ __forceinline__ vf4 quant4(vf4 v, int up, int dn, bool passthru) {
  vf4 r;
#pragma unroll
  for (int k = 0; k < 4; ++k) {
    float t = rintf(ldexpf(v[k], up));                      // v_ldexp_f32 + v_rndne_f32 (RNE)
    t = fminf(127.0f, fmaxf(-128.0f, t));
    r[k] = passthru ? v[k] : ldexpf(t, dn);
  }
  return r;
}

__global__ __launch_bounds__(THREADS)
void swalp_quant(const vf4* __restrict__ x4, vf4* __restrict__ o4,
                 const float* __restrict__ x, float* __restrict__ o,
                 unsigned n4, unsigned n, const unsigned* __restrict__ ws) {
  const unsigned mb = *ws;                    // uniform -> s_load_b32
  const bool passthru = (mb == 0u);
  int e = 0;
  if (!passthru) {
    e = ilogbf(__uint_as_float(mb));          // exact floor(log2(maxabs)), denorm-safe
    e = e < -128 ? -128 : (e > 127 ? 127 : e);
  }
  const int up = 6 - e;                       // bits-2 = 6 for 8-bit
  const int dn = e - 6;

  const unsigned stride = gridDim.x * THREADS;
  unsigned i = blockIdx.x * THREADS + threadIdx.x;
  for (; i + stride < n4; i += 2u * stride) {
    vf4 a = x4[i];                                          // global_load_b128
    vf4 b = x4[i + stride];
    __builtin_nontemporal_store(quant4(a, up, dn, passthru), &o4[i]);        // b128 TH=NT
    __builtin_nontemporal_store(quant4(b, up, dn, passthru), &o4[i + stride]);
  }
  if (i < n4)
    __builtin_nontemporal_store(quant4(x4[i], up, dn, passthru), &o4[i]);
  if (blockIdx.x == 0 && threadIdx.x == 0)                  // tail (n % 4), generic
    for (unsigned t = n4 * 4u; t < n; ++t) {
      float v = x[t];
      float q = fminf(127.0f, fmaxf(-128.0f, rintf(ldexpf(v, up))));
      float r = passthru ? v : ldexpf(q, dn);
      __builtin_nontemporal_store(r, &o[t]);
    }
}

extern "C" void kernel_launch(void* const* d_in, const int* in_sizes, int n_in,
                              void* d_out, int out_size, void* d_ws, size_t ws_size,
                              hipStream_t stream) {
  const float* x  = (const float*)d_in[0];
  float*       ov = (float*)d_out;
  unsigned*    ws = (unsigned*)d_ws;

  const unsigned n  = (unsigned)in_sizes[0];
  const unsigned n4 = n >> 2;

  unsigned b = (n4 + THREADS - 1) / THREADS;
  int blocks = (int)(b > MAXBLOCKS ? MAXBLOCKS : (b < 1u ? 1u : b));

  swalp_init<<<1, 1, 0, stream>>>(ws);   // ws is poisoned by harness; re-zero every call
  swalp_absmax<<<blocks, THREADS, 0, stream>>>((const vf4*)x, x, n4, n, ws);
  swalp_quant <<<blocks, THREADS, 0, stream>>>((const vf4*)x, (vf4*)ov, x, ov, n4, n, ws);
}